// SplineLayer_19086834663690
// MI455X (gfx1250) — compile-verified
//
#include <hip/hip_runtime.h>
#include <math.h>
#include <stdint.h>

#define IN_DIM    256
#define OUT_DIM   128
#define PHI_KNOTS 300
#define PSI_KNOTS 200
#define ROWS_PER_BLOCK 8

// float offsets into d_ws (tables stored interleaved as float2 {A,B})
#define WS_PHI 0      // 299 float2 = 598 floats
#define WS_PSI 640    // 199 float2 = 398 floats

// ---------------------------------------------------------------------------
// Prep: sort phi coeffs (reference does jnp.sort) and build per-interval
// {intercept, slope} float2 tables so each spline eval is one b64 gather +
// one FMA:  val = A[idx] + B[idx] * v.
// ---------------------------------------------------------------------------
__global__ __launch_bounds__(512)
void spline_prep_kernel(const float* __restrict__ phi_coeffs,
                        const float* __restrict__ psi_coeffs,
                        const float* __restrict__ phi_knots,
                        const float* __restrict__ psi_knots,
                        float* __restrict__ ws)
{
    __shared__ float s[512];
    const int tid = threadIdx.x;
    s[tid] = (tid < PHI_KNOTS) ? phi_coeffs[tid] : INFINITY;
    __syncthreads();

    // Bitonic sort (ascending) of 512 elements (300 real + inf padding).
    for (int k = 2; k <= 512; k <<= 1) {
        for (int j = k >> 1; j > 0; j >>= 1) {
            int ixj = tid ^ j;
            if (ixj > tid) {
                float a = s[tid], b = s[ixj];
                bool up = ((tid & k) == 0);
                if ((a > b) == up) { s[tid] = b; s[ixj] = a; }
            }
            __syncthreads();
        }
    }

    if (tid < PHI_KNOTS - 1) {
        float k0 = phi_knots[tid], k1 = phi_knots[tid + 1];
        float c0 = s[tid],         c1 = s[tid + 1];
        float slope = (c1 - c0) / (k1 - k0);
        ((float2*)(ws + WS_PHI))[tid] = make_float2(fmaf(-slope, k0, c0), slope);
    }
    if (tid < PSI_KNOTS - 1) {
        float k0 = psi_knots[tid], k1 = psi_knots[tid + 1];
        float c0 = psi_coeffs[tid], c1 = psi_coeffs[tid + 1];
        float slope = (c1 - c0) / (k1 - k0);
        ((float2*)(ws + WS_PSI))[tid] = make_float2(fmaf(-slope, k0, c0), slope);
    }
}

// ---------------------------------------------------------------------------
// Main: one wave32 per batch row, 4 outputs per lane (o = lane + 32k).
// Per output-term: 1 ds_load_b64 gather + 1/4 ds_load_b64 broadcast + FMAs.
// Tables async-staged to LDS via CDNA5 global_load_async_to_lds (ASYNCcnt).
// ---------------------------------------------------------------------------
__global__ __launch_bounds__(256)
void spline_main_kernel(const float* __restrict__ x,
                        const float* __restrict__ lambdas,
                        const float* __restrict__ eta_p,
                        const float* __restrict__ ws,
                        float* __restrict__ out,
                        int B)
{
    __shared__ float2 shPhi[PHI_KNOTS - 1];               // {A,B} per interval
    __shared__ float2 shPsi[PSI_KNOTS - 1];
    __shared__ float2 shXL[ROWS_PER_BLOCK * IN_DIM];      // {x_i, lambda_i}

    const int tid = threadIdx.x;
    const int b0  = blockIdx.x * ROWS_PER_BLOCK;

    // --- async-stage interleaved spline tables (contiguous in ws) ---
    {
        const float* gphi = ws + WS_PHI;
        unsigned ldsPhi = (unsigned)(uintptr_t)(&shPhi[0]);
        {
            const float* g0 = gphi + 2 * tid;
            unsigned l0 = ldsPhi + (unsigned)(tid * 8);
            asm volatile("global_load_async_to_lds_b64 %0, %1, off"
                         :: "v"(l0), "v"(g0) : "memory");
        }
        if (tid < (PHI_KNOTS - 1) - 256) {                // entries 256..298
            const float* g1 = gphi + 2 * (tid + 256);
            unsigned l1 = ldsPhi + (unsigned)((tid + 256) * 8);
            asm volatile("global_load_async_to_lds_b64 %0, %1, off"
                         :: "v"(l1), "v"(g1) : "memory");
        }
        if (tid < PSI_KNOTS - 1) {
            const float* g2 = ws + WS_PSI + 2 * tid;
            unsigned l2 = (unsigned)(uintptr_t)(&shPsi[0]) + (unsigned)(tid * 8);
            asm volatile("global_load_async_to_lds_b64 %0, %1, off"
                         :: "v"(l2), "v"(g2) : "memory");
        }
    }

    // --- interleave x rows with lambdas into LDS ---
    {
        float lv = lambdas[tid];
        const float* xg = x + (size_t)b0 * IN_DIM + tid;
        #pragma unroll
        for (int r = 0; r < ROWS_PER_BLOCK; ++r) {
            if (b0 + r < B)
                shXL[r * IN_DIM + tid] = make_float2(xg[(size_t)r * IN_DIM], lv);
        }
    }

    asm volatile("s_wait_asynccnt 0" ::: "memory");
    __syncthreads();

    const int lane = tid & 31;
    const int w    = tid >> 5;            // wave id == row within block
    const int b    = b0 + w;
    if (b >= B) return;

    const float eta = eta_p[0];
    const float2* __restrict__ xl = &shXL[w * IN_DIM];

    const float q0 = (float)lane,        q1 = (float)(lane + 32);
    const float q2 = (float)(lane + 64), q3 = (float)(lane + 96);
    const float sq0 = eta * q0, sq1 = eta * q1, sq2 = eta * q2, sq3 = eta * q3;
    const float phiScale = (float)(PHI_KNOTS - 1);        // phi range [0,1]

    float a0 = 0.f, a1 = 0.f, a2 = 0.f, a3 = 0.f;

    #pragma unroll 2
    for (int i = 0; i < IN_DIM; ++i) {
        const float2 v = xl[i];           // one b64 broadcast / 4 terms
        const float xi = v.x, li = v.y;

        float s0 = fminf(fmaxf(xi + sq0, 0.f), 1.f);
        float s1 = fminf(fmaxf(xi + sq1, 0.f), 1.f);
        float s2 = fminf(fmaxf(xi + sq2, 0.f), 1.f);
        float s3 = fminf(fmaxf(xi + sq3, 0.f), 1.f);

        int i0 = (int)(s0 * phiScale); i0 = i0 > PHI_KNOTS - 2 ? PHI_KNOTS - 2 : i0;
        int i1 = (int)(s1 * phiScale); i1 = i1 > PHI_KNOTS - 2 ? PHI_KNOTS - 2 : i1;
        int i2 = (int)(s2 * phiScale); i2 = i2 > PHI_KNOTS - 2 ? PHI_KNOTS - 2 : i2;
        int i3 = (int)(s3 * phiScale); i3 = i3 > PHI_KNOTS - 2 ? PHI_KNOTS - 2 : i3;

        float2 t0 = shPhi[i0];
        float2 t1 = shPhi[i1];
        float2 t2 = shPhi[i2];
        float2 t3 = shPhi[i3];

        a0 = fmaf(li, fmaf(t0.y, s0, t0.x), a0);
        a1 = fmaf(li, fmaf(t1.y, s1, t1.x), a1);
        a2 = fmaf(li, fmaf(t2.y, s2, t2.x), a2);
        a3 = fmaf(li, fmaf(t3.y, s3, t3.x), a3);
    }

    // --- psi spline + store (4 coalesced b32 stores per lane) ---
    const float psiScale = (float)(PSI_KNOTS - 1) / 22.0f;
    float* __restrict__ orow = out + (size_t)b * OUT_DIM;

    float accs[4] = {a0, a1, a2, a3};
    float qs[4]   = {q0, q1, q2, q3};
    #pragma unroll
    for (int k = 0; k < 4; ++k) {
        float inner = fminf(fmaxf(accs[k] + qs[k], -10.0f), 12.0f);
        int p = (int)((inner + 10.0f) * psiScale);
        p = p > PSI_KNOTS - 2 ? PSI_KNOTS - 2 : p;
        float2 ab = shPsi[p];
        orow[lane + 32 * k] = fmaf(ab.y, inner, ab.x);
    }
}

// ---------------------------------------------------------------------------
extern "C" void kernel_launch(void* const* d_in, const int* in_sizes, int n_in,
                              void* d_out, int out_size, void* d_ws, size_t ws_size,
                              hipStream_t stream) {
    const float* x          = (const float*)d_in[0];
    const float* phi_coeffs = (const float*)d_in[1];
    const float* psi_coeffs = (const float*)d_in[2];
    const float* lambdas    = (const float*)d_in[3];
    const float* eta        = (const float*)d_in[4];
    const float* phi_knots  = (const float*)d_in[5];
    const float* psi_knots  = (const float*)d_in[6];
    float* out = (float*)d_out;
    float* ws  = (float*)d_ws;

    const int B = in_sizes[0] / IN_DIM;   // 2048
    const int nblk = (B + ROWS_PER_BLOCK - 1) / ROWS_PER_BLOCK;

    spline_prep_kernel<<<1, 512, 0, stream>>>(phi_coeffs, psi_coeffs,
                                              phi_knots, psi_knots, ws);
    spline_main_kernel<<<nblk, 256, 0, stream>>>(x, lambdas, eta, ws, out, B);
}